// PoolKDropout_39496519254068
// MI455X (gfx1250) — compile-verified
//
#include <hip/hip_runtime.h>
#include <stdint.h>

#define THREADS   256
#define D_COLS    4096u        // feature dim
#define M_ROWS    2048u        // masks per batch (pool block rows)
#define APPLY_BLOCKS 2048u

// clang-native float4 (accepted by __builtin_nontemporal_store, still lowers to b128 ops)
typedef __attribute__((ext_vector_type(4))) float v4f;

// ---------------- threefry-2x32 (matches jax.random) ----------------
__device__ __forceinline__ uint32_t rotl32(uint32_t x, uint32_t r) {
  return (x << r) | (x >> (32u - r));
}

__device__ __forceinline__ void tf_round4(uint32_t& x0, uint32_t& x1,
                                          uint32_t ra, uint32_t rb,
                                          uint32_t rc, uint32_t rd) {
  x0 += x1; x1 = rotl32(x1, ra); x1 ^= x0;
  x0 += x1; x1 = rotl32(x1, rb); x1 ^= x0;
  x0 += x1; x1 = rotl32(x1, rc); x1 ^= x0;
  x0 += x1; x1 = rotl32(x1, rd); x1 ^= x0;
}

__device__ __forceinline__ void threefry2x32(uint32_t k0, uint32_t k1,
                                             uint32_t c0, uint32_t c1,
                                             uint32_t& o0, uint32_t& o1) {
  uint32_t k2 = k0 ^ k1 ^ 0x1BD11BDAu;
  uint32_t x0 = c0 + k0, x1 = c1 + k1;
  tf_round4(x0, x1, 13, 15, 26, 6);  x0 += k1; x1 += k2 + 1u;
  tf_round4(x0, x1, 17, 29, 16, 24); x0 += k2; x1 += k0 + 2u;
  tf_round4(x0, x1, 13, 15, 26, 6);  x0 += k0; x1 += k1 + 3u;
  tf_round4(x0, x1, 17, 29, 16, 24); x0 += k1; x1 += k2 + 4u;
  tf_round4(x0, x1, 13, 15, 26, 6);  x0 += k2; x1 += k0 + 5u;
  o0 = x0; o1 = x1;
}

// ---------------- kernel A: build 1 MiB bit-mask block in scratch ----------------
// mask_block layout: [M_ROWS][128] uint32 words; word w of a row covers cols 32w..32w+31.
// JAX counter mapping: bits[col]       = out0 of threefry(fk, (col, col+2048))   col <  2048
//                      bits[col+2048]  = out1 of the same call.
__global__ void __launch_bounds__(THREADS)
poolk_dropout_maskgen(const int* __restrict__ seed_idxs,
                      uint32_t* __restrict__ maskws) {
  uint32_t t   = blockIdx.x * THREADS + threadIdx.x;   // 0 .. M_ROWS*64-1
  uint32_t row = t >> 6;
  uint32_t w   = t & 63u;

  uint32_t idx = (uint32_t)seed_idxs[row];

  // fold_in(key(42), idx) == threefry2x32(key=(0,42), ctr=(0, idx))
  uint32_t fk0, fk1;
  threefry2x32(0u, 42u, 0u, idx, fk0, fk1);

  uint32_t lo = 0u, hi = 0u;
  uint32_t base = w << 5;                               // counter base = 32*w
  #pragma unroll 4
  for (uint32_t b = 0; b < 32u; ++b) {
    uint32_t o0, o1;
    threefry2x32(fk0, fk1, base + b, base + b + M_ROWS, o0, o1);
    // uniform >= 0.5  <=>  top bit of the 32 random bits is set
    lo |= (o0 >> 31) << b;
    hi |= (o1 >> 31) << b;
  }
  maskws[(row << 7) + w]       = lo;   // cols 32w .. 32w+31
  maskws[(row << 7) + 64u + w] = hi;   // cols 2048+32w .. 2048+32w+31
}

// ---------------- kernel B: streaming apply via async-to-LDS double buffer ----------------
__global__ void __launch_bounds__(THREADS)
poolk_dropout_apply(const float* __restrict__ x,
                    const uint32_t* __restrict__ maskws,
                    float* __restrict__ out,
                    unsigned n4) {
  __shared__ v4f stage[2 * THREADS];             // two 4 KiB buffers, one slot/thread

  const unsigned tid    = threadIdx.x;
  const unsigned g0     = blockIdx.x * THREADS + tid;  // float4 index
  const unsigned stride = gridDim.x * THREADS;
  const unsigned iters  = n4 / stride;                 // exact by construction (16)

  // raw LDS byte addresses for the async DMA destination (low 32 bits of flat addr)
  const uint32_t lds0 = (uint32_t)(uintptr_t)&stage[tid];
  const uint32_t lds1 = (uint32_t)(uintptr_t)&stage[THREADS + tid];

  // prologue: fill buffer 0
  asm volatile("global_load_async_to_lds_b128 %0, %1, %2"
               :: "v"(lds0), "v"(g0 * 16u), "s"(x) : "memory");

  for (unsigned i = 0; i < iters; ++i) {
    const unsigned gi = g0 + i * stride;

    if (i + 1u < iters) {
      const uint32_t nxt = (i & 1u) ? lds0 : lds1;
      asm volatile("global_load_async_to_lds_b128 %0, %1, %2"
                   :: "v"(nxt), "v"((gi + stride) * 16u), "s"(x) : "memory");
      // two loads outstanding; oldest (current buffer) is done when <=1 remain
      asm volatile("s_wait_asynccnt 1" ::: "memory");
    } else {
      asm volatile("s_wait_asynccnt 0" ::: "memory");
    }

    const v4f v = stage[(i & 1u) * THREADS + tid];      // ds_load_b128

    // mask lookup: row = gi / (4096/4); 4 cols of this float4 share one mask word
    const uint32_t row = gi >> 10;
    const uint32_t mw  = maskws[((row & (M_ROWS - 1u)) << 7) + ((gi & 1023u) >> 3)];
    const uint32_t b   = (gi & 7u) << 2;

    v4f r;
    r.x = v.x * (((mw >> (b + 0u)) & 1u) ? 2.0f : 0.0f);
    r.y = v.y * (((mw >> (b + 1u)) & 1u) ? 2.0f : 0.0f);
    r.z = v.z * (((mw >> (b + 2u)) & 1u) ? 2.0f : 0.0f);
    r.w = v.w * (((mw >> (b + 3u)) & 1u) ? 2.0f : 0.0f);

    // output is written once and never re-read: non-temporal store
    __builtin_nontemporal_store(r, (v4f*)(out) + gi);
  }
}

extern "C" void kernel_launch(void* const* d_in, const int* in_sizes, int n_in,
                              void* d_out, int out_size, void* d_ws, size_t ws_size,
                              hipStream_t stream) {
  const float* x         = (const float*)d_in[0];
  const int*   seed_idxs = (const int*)d_in[1];
  float*       out       = (float*)d_out;
  uint32_t*    maskws    = (uint32_t*)d_ws;            // needs M_ROWS*128*4 = 1 MiB

  const unsigned m  = (unsigned)in_sizes[1];           // 2048
  const unsigned n4 = (unsigned)out_size / 4u;         // 8,388,608 float4 groups

  // Kernel A: 2048 rows * 64 word-pairs = 131072 threads
  const unsigned blocksA = (m * 64u) / THREADS;        // 512
  poolk_dropout_maskgen<<<blocksA, THREADS, 0, stream>>>(seed_idxs, maskws);

  // Kernel B: 2048 blocks * 256 threads, 16 iterations each (exact cover)
  poolk_dropout_apply<<<APPLY_BLOCKS, THREADS, 0, stream>>>(x, maskws, out, n4);
}